// GraceGCN_78039555768418
// MI455X (gfx1250) — compile-verified
//
#include <hip/hip_runtime.h>

typedef __attribute__((ext_vector_type(2))) float v2f;
typedef __attribute__((ext_vector_type(8))) float v8f;

// ---------------------------------------------------------------------------
// Degree accumulation: deg_out[src[e]] += 1, deg_in[dst[e]] += 1 (fp32 HW atomics)
// ---------------------------------------------------------------------------
__global__ void gcn_deg_kernel(const int* __restrict__ src, const int* __restrict__ dst,
                               float* __restrict__ deg_out, float* __restrict__ deg_in,
                               int nE) {
    int e = blockIdx.x * blockDim.x + threadIdx.x;
    if (e < nE) {
        unsafeAtomicAdd(&deg_out[src[e]], 1.0f);
        unsafeAtomicAdd(&deg_in[dst[e]], 1.0f);
    }
}

// v[i] = rsqrt(max(v[i], 1))  (in place, covers both deg arrays at once)
__global__ void gcn_rsqrt_kernel(float* __restrict__ v, int n) {
    int i = blockIdx.x * blockDim.x + threadIdx.x;
    if (i < n) {
        float d = v[i];
        d = d < 1.0f ? 1.0f : d;
        v[i] = rsqrtf(d);
    }
}

// ---------------------------------------------------------------------------
// Tiled fp32 GEMM via V_WMMA_F32_16X16X4_F32, one wave32 per 16x16 C tile.
//   C[r, c] = (Aeff[r, :] @ B[:, c]) * s_row[r]
// PREACT: Aeff[m,k] = relu(A[m,k] * s_in[m] + bias[k])   (fused layer-1 epilogue)
// A row-major [M,K], B row-major [K,Ncols], C row-major [M,Ncols].
// ---------------------------------------------------------------------------
template <bool PREACT>
__global__ void gcn_gemm_wmma(const float* __restrict__ A, const float* __restrict__ B,
                              float* __restrict__ C, const float* __restrict__ s_row,
                              const float* __restrict__ s_in, const float* __restrict__ bias,
                              int K, int Ncols, int colTiles, int totalTiles, int Nrows) {
    int wave = blockIdx.x * (blockDim.x >> 5) + (threadIdx.x >> 5);
    if (wave >= totalTiles) return;                 // wave-uniform: EXEC stays full
    int tileM = wave / colTiles;
    int tileN = wave - tileM * colTiles;
    int lane  = threadIdx.x & 31;
    int mn    = lane & 15;                          // m for A, n for B/C
    int kb    = (lane >> 4) << 1;                   // K sub-pair: 0 or 2

    int row = tileM * 16 + mn;
    if (row >= Nrows) row = Nrows - 1;              // defensive clamp (reads only)
    int col = tileN * 16 + mn;

    const float* arow = A + (size_t)row * K;
    float si = PREACT ? s_in[row] : 0.0f;

    v8f acc = {};
    for (int k0 = 0; k0 < K; k0 += 4) {
        int kk = k0 + kb;
        float2 av = *(const float2*)(arow + kk);    // 8B-aligned: kk is even
        if (PREACT) {
            av.x = fmaxf(av.x * si + bias[kk],     0.0f);
            av.y = fmaxf(av.y * si + bias[kk + 1], 0.0f);
        }
        v2f a; a.x = av.x; a.y = av.y;
        v2f b;
        b.x = B[(size_t)kk       * Ncols + col];
        b.y = B[(size_t)(kk + 1) * Ncols + col];
        // D = A(16x4, f32) * B(4x16, f32) + C : exact fp32, 8-arg form
        acc = __builtin_amdgcn_wmma_f32_16x16x4_f32(
            /*neg_a=*/false, a, /*neg_b=*/false, b,
            /*c_mod=*/(short)0, acc, /*reuse_a=*/false, /*reuse_b=*/false);
    }

    int mbase = tileM * 16 + ((lane >> 4) << 3);
    for (int i = 0; i < 8; ++i) {
        int r = mbase + i;
        if (r < Nrows)
            C[(size_t)r * Ncols + col] = acc[i] * s_row[r];
    }
}

// ---------------------------------------------------------------------------
// Edge scatter-add: agg[dst[e], :] += y[src[e], :]
// One thread per (edge, 4-float chunk); float4 gather + 4 fp32 HW atomics.
// ---------------------------------------------------------------------------
__global__ void gcn_edge_scatter(const int* __restrict__ src, const int* __restrict__ dst,
                                 const float* __restrict__ y, float* __restrict__ agg,
                                 int nE, int D, int logChunks) {
    long long gid = (long long)blockIdx.x * blockDim.x + threadIdx.x;
    int e = (int)(gid >> logChunks);
    if (e >= nE) return;
    int c = ((int)gid & ((1 << logChunks) - 1)) << 2;

    int s = src[e], d = dst[e];
    float4 v = *(const float4*)(y + (size_t)s * D + c);
    float* p = agg + (size_t)d * D + c;
    unsafeAtomicAdd(p + 0, v.x);
    unsafeAtomicAdd(p + 1, v.y);
    unsafeAtomicAdd(p + 2, v.z);
    unsafeAtomicAdd(p + 3, v.w);
}

// out[i] = relu(agg[i] * s_in[i / D] + bias[i % D])
__global__ void gcn_finalize(const float* __restrict__ agg, const float* __restrict__ s_in,
                             const float* __restrict__ bias, float* __restrict__ out,
                             int total, int logD) {
    int i = blockIdx.x * blockDim.x + threadIdx.x;
    if (i < total) {
        int r = i >> logD;
        int d = i & ((1 << logD) - 1);
        out[i] = fmaxf(agg[i] * s_in[r] + bias[d], 0.0f);
    }
}

// ---------------------------------------------------------------------------
extern "C" void kernel_launch(void* const* d_in, const int* in_sizes, int n_in,
                              void* d_out, int out_size, void* d_ws, size_t ws_size,
                              hipStream_t stream) {
    const float* features = (const float*)d_in[0];
    const int*   edges    = (const int*)  d_in[1];
    const float* W1       = (const float*)d_in[2];
    const float* b1       = (const float*)d_in[3];
    const float* W2       = (const float*)d_in[4];
    const float* b2       = (const float*)d_in[5];
    float*       out      = (float*)d_out;

    const int IN = 128, HID = 128, OUT = 64;
    const int N = in_sizes[0] / IN;
    const int E = in_sizes[1] / 2;
    const int* src = edges;
    const int* dst = edges + E;

    float* ws    = (float*)d_ws;
    float* s_out = ws;                          // N  (deg_out -> rsqrt scale)
    float* s_in  = ws + N;                      // N  (deg_in  -> rsqrt scale)
    float* bufA  = ws + 2 * (size_t)N;          // N*HID : y1, then y2 (N*OUT)
    float* bufB  = bufA + (size_t)N * HID;      // N*HID : agg1, then agg2 (N*OUT)

    // --- degrees -> rsqrt scales ---
    hipMemsetAsync(s_out, 0, 2 * (size_t)N * sizeof(float), stream);
    gcn_deg_kernel<<<(E + 255) / 256, 256, 0, stream>>>(src, dst, s_out, s_in, E);
    gcn_rsqrt_kernel<<<(2 * N + 255) / 256, 256, 0, stream>>>(s_out, 2 * N);

    // --- layer 1 GEMM: bufA = (features @ W1) * s_out  (matmul-first reorder) ---
    {
        int colTiles = HID / 16, rowTiles = (N + 15) / 16;
        int total = rowTiles * colTiles;
        gcn_gemm_wmma<false><<<(total + 7) / 8, 256, 0, stream>>>(
            features, W1, bufA, s_out, nullptr, nullptr, IN, HID, colTiles, total, N);
    }

    // --- layer 1 scatter: bufB[dst] += bufA[src]  (128 wide) ---
    hipMemsetAsync(bufB, 0, (size_t)N * HID * sizeof(float), stream);
    {
        long long threads = (long long)E * (HID / 4);
        gcn_edge_scatter<<<(int)((threads + 255) / 256), 256, 0, stream>>>(
            src, dst, bufA, bufB, E, HID, 5);
    }

    // --- layer 2 GEMM (fused layer-1 epilogue in A-load):
    //     bufA = (relu(bufB * s_in + b1) @ W2) * s_out   (64 wide) ---
    {
        int colTiles = OUT / 16, rowTiles = (N + 15) / 16;
        int total = rowTiles * colTiles;
        gcn_gemm_wmma<true><<<(total + 7) / 8, 256, 0, stream>>>(
            bufB, W2, bufA, s_out, s_in, b1, HID, OUT, colTiles, total, N);
    }

    // --- layer 2 scatter: bufB[dst] += bufA[src]  (64 wide, half the traffic) ---
    hipMemsetAsync(bufB, 0, (size_t)N * OUT * sizeof(float), stream);
    {
        long long threads = (long long)E * (OUT / 4);
        gcn_edge_scatter<<<(int)((threads + 255) / 256), 256, 0, stream>>>(
            src, dst, bufA, bufB, E, OUT, 4);
    }

    // --- final: out = relu(bufB * s_in + b2) ---
    gcn_finalize<<<(N * OUT + 255) / 256, 256, 0, stream>>>(
        bufB, s_in, b2, out, N * OUT, 6);
}